// ExpansionContrastModule_7352984011485
// MI455X (gfx1250) — compile-verified
//
#include <hip/hip_runtime.h>

typedef __attribute__((ext_vector_type(2))) float v2f;
typedef __attribute__((ext_vector_type(4))) float f4;
typedef __attribute__((ext_vector_type(8))) float v8f;

#define IMGW 256
#define HW   65536          // 256*256
#define TILE 32
#define PADW 38             // TILE + 2*3 halo
#define NPIX (PADW * PADW)  // 1444
#define NGRP ((NPIX + 15) / 16)  // 91 groups of 16 pixels
#define XSTR 20             // LDS per-pixel channel stride (16-B aligned)

__device__ __forceinline__ float sigm(float x) { return 1.0f / (1.0f + __expf(-x)); }
__device__ __forceinline__ f4 min4(f4 a, f4 b) {
  f4 r; r[0]=fminf(a[0],b[0]); r[1]=fminf(a[1],b[1]); r[2]=fminf(a[2],b[2]); r[3]=fminf(a[3],b[3]); return r;
}
__device__ __forceinline__ f4 max4(f4 a, f4 b) {
  f4 r; r[0]=fmaxf(a[0],b[0]); r[1]=fmaxf(a[1],b[1]); r[2]=fmaxf(a[2],b[2]); r[3]=fmaxf(a[3],b[3]); return r;
}

__global__ __launch_bounds__(256) void ecm_fused(
    const float* __restrict__ cen,  const float* __restrict__ mas,
    const float* __restrict__ w_in, const float* __restrict__ b_in,
    const float* __restrict__ scale1, const float* __restrict__ scale2,
    const float* __restrict__ scale3, const float* __restrict__ bc_w,
    const float* __restrict__ bn_g, const float* __restrict__ bn_b,
    const float* __restrict__ bn_m, const float* __restrict__ bn_v,
    const float* __restrict__ w_out, const float* __restrict__ b_out,
    const float* __restrict__ mas_w1, const float* __restrict__ mas_b1,
    const float* __restrict__ mas_w2, const float* __restrict__ mas_b2,
    float* __restrict__ out)
{
  __shared__ float xs[NPIX * XSTR];   // x on padded tile: [pixel][16ch(+pad)]

  const int tid  = threadIdx.x;
  const int lane = tid & 31;
  const int wv   = tid >> 5;
  const int tx0  = blockIdx.x * TILE;
  const int ty0  = blockIdx.y * TILE;
  const int bb_  = blockIdx.z;        // batch

  // ---------------- Phase 1: x = w_in * cen + b_in via V_WMMA_F32_16X16X4_F32 ----------------
  // A layout (16x4 f32): lanes 0-15 -> M=lane, K={k0+0,k0+1}; lanes 16-31 -> M=lane-16, K={k0+2,k0+3}
  const int m  = lane & 15;
  const int kh = (lane >> 4) << 1;    // 0 or 2

  v2f A[32];
  #pragma unroll
  for (int s = 0; s < 32; ++s) {
    A[s][0] = w_in[m * 128 + 4 * s + kh];
    A[s][1] = w_in[m * 128 + 4 * s + kh + 1];
  }

  for (int g = wv; g < NGRP; g += 8) {        // wave-uniform loop: EXEC all-ones at WMMA
    const int pp = g * 16 + (lane & 15);      // this lane's pixel (B column N = lane%16)
    const int py = pp / PADW;
    const int px = pp - py * PADW;
    const int iy = ty0 + py - 3;
    const int ix = tx0 + px - 3;
    const bool inimg = (pp < NPIX) && (iy >= 0) && (iy < IMGW) && (ix >= 0) && (ix < IMGW);

    // Clamp dead lanes to a safe in-bounds address and load UNconditionally:
    // WMMA column n depends only on B column n, and invalid pixels are
    // overwritten with zeros at the LDS store, so garbage columns are harmless.
    const int off = inimg ? (iy * IMGW + ix) : 0;
    const float* pc = cen + (size_t)(bb_ * 128 + kh) * HW + off;

    v2f Bv[32];
    #pragma unroll
    for (int s = 0; s < 32; ++s) {
      Bv[s][0] = pc[0];
      Bv[s][1] = pc[HW];
      pc += 4 * HW;
    }

    v8f acc = {0.f,0.f,0.f,0.f,0.f,0.f,0.f,0.f};
    #pragma unroll
    for (int s = 0; s < 32; ++s)
      acc = __builtin_amdgcn_wmma_f32_16x16x4_f32(
          false, A[s], false, Bv[s], (short)0, acc, false, false);

    if (pp < NPIX) {
      const int chb = (lane >> 4) << 3;       // lanes<16: ch 0..7, lanes>=16: ch 8..15
      float* dst = &xs[pp * XSTR + chb];
      if (inimg) {
        f4 lo = { acc[0] + b_in[chb+0], acc[1] + b_in[chb+1],
                  acc[2] + b_in[chb+2], acc[3] + b_in[chb+3] };
        f4 hi = { acc[4] + b_in[chb+4], acc[5] + b_in[chb+5],
                  acc[6] + b_in[chb+6], acc[7] + b_in[chb+7] };
        *(f4*)dst = lo; *(f4*)(dst + 4) = hi;
      } else {                                 // conv zero-padding: x == 0 outside image
        f4 z = {0.f,0.f,0.f,0.f};
        *(f4*)dst = z; *(f4*)(dst + 4) = z;
      }
    }
  }
  __syncthreads();

  // ---------------- Phase 2: stencils + gate + final multiply ----------------
  float s1a = scale1[0], s1b = scale1[1], s1c = scale1[2];
  { float mx = fmaxf(s1a, fmaxf(s1b, s1c));
    float ea = __expf(s1a-mx), eb = __expf(s1b-mx), ec = __expf(s1c-mx);
    float inv = 1.f/(ea+eb+ec); s1a = ea*inv; s1b = eb*inv; s1c = ec*inv; }
  float s2a = scale2[0], s2b = scale2[1], s2c = scale2[2];
  { float mx = fmaxf(s2a, fmaxf(s2b, s2c));
    float ea = __expf(s2a-mx), eb = __expf(s2b-mx), ec = __expf(s2c-mx);
    float inv = 1.f/(ea+eb+ec); s2a = ea*inv; s2b = eb*inv; s2c = ec*inv; }
  float s3a = scale3[0], s3b = scale3[1], s3c = scale3[2], s3d = scale3[3];
  { float mx = fmaxf(fmaxf(s3a, s3b), fmaxf(s3c, s3d));
    float ea = __expf(s3a-mx), eb = __expf(s3b-mx), ec = __expf(s3c-mx), ed = __expf(s3d-mx);
    float inv = 1.f/(ea+eb+ec+ed); s3a = ea*inv; s3b = eb*inv; s3c = ec*inv; s3d = ed*inv; }

  float bnA[16], bnB[16], wo[16];
  #pragma unroll
  for (int c = 0; c < 16; ++c) {
    float inv = bn_g[c] * rsqrtf(bn_v[c] + 1e-5f);
    bnA[c] = inv;
    bnB[c] = bn_b[c] - bn_m[c] * inv;
    wo[c]  = w_out[c];
  }
  float mw[9];
  #pragma unroll
  for (int k = 0; k < 9; ++k) mw[k] = mas_w1[k];
  const float mb1 = mas_b1[0], mw2 = mas_w2[0], mb2 = mas_b2[0], bo = b_out[0];

  for (int i = 0; i < 4; ++i) {
    const int p  = tid + (i << 8);
    const int y  = p >> 5, x = p & 31;
    const float* c0 = &xs[((y + 3) * PADW + (x + 3)) * XSTR];

    float yv[16];
    #pragma unroll
    for (int cc = 0; cc < 16; cc += 4) {
      const f4 xc = *(const f4*)(c0 + cc);
      f4 rr[2];
      #pragma unroll
      for (int si = 0; si < 2; ++si) {
        const int s  = si ? 3 : 1;
        const int du = -s * PADW * XSTR, dd = s * PADW * XSTR;
        const int dl = -s * XSTR,        dr = s * XSTR;
        const f4 nUL = *(const f4*)(c0 + cc + du + dl);
        const f4 nU  = *(const f4*)(c0 + cc + du);
        const f4 nUR = *(const f4*)(c0 + cc + du + dr);
        const f4 nL  = *(const f4*)(c0 + cc + dl);
        const f4 nR  = *(const f4*)(c0 + cc + dr);
        const f4 nDL = *(const f4*)(c0 + cc + dd + dl);
        const f4 nD  = *(const f4*)(c0 + cc + dd);
        const f4 nDR = *(const f4*)(c0 + cc + dd + dr);
        const f4 o0 = (xc - nUL) * (xc - nDR);
        const f4 o1 = (xc - nU ) * (xc - nD );
        const f4 o2 = (xc - nUR) * (xc - nDL);
        const f4 o3 = (xc - nL ) * (xc - nR );
        const f4 mn = min4(min4(o0, o1), min4(o2, o3));
        const f4 mx = max4(max4(o0, o1), max4(o2, o3));
        const f4 me = (o0 + o1 + o2 + o3) * 0.25f;
        rr[si] = mn * s1a + me * s1b + mx * s1c;
      }
      const f4 mn = min4(rr[0], rr[1]);
      const f4 mx = max4(rr[0], rr[1]);
      const f4 me = (rr[0] + rr[1]) * 0.5f;
      const f4 yc = mn * s2a + me * s2b + mx * s2c;
      yv[cc+0] = yc[0]; yv[cc+1] = yc[1]; yv[cc+2] = yc[2]; yv[cc+3] = yc[3];
    }

    // 16x16 matvec + BN + SiLU + 16->1 projection (weights are uniform -> scalar loads)
    float os = bo;
    #pragma unroll
    for (int o = 0; o < 16; ++o) {
      float a = 0.f;
      #pragma unroll
      for (int c = 0; c < 16; ++c) a = fmaf(bc_w[o * 16 + c], yv[c], a);
      a = fmaf(a, bnA[o], bnB[o]);
      a = a * sigm(a);                       // SiLU
      os = fmaf(wo[o], a, os);
    }
    const float om = sigm(os);

    // mask branch: 3x3 conv (zero pad) + SiLU + 1x1 + sigmoid
    const int iy = ty0 + y, ix = tx0 + x;
    float macc = mb1;
    #pragma unroll
    for (int ky = 0; ky < 3; ++ky)
      #pragma unroll
      for (int kx = 0; kx < 3; ++kx) {
        const int yy = iy + ky - 1, xx = ix + kx - 1;
        float v = 0.f;
        if (yy >= 0 && yy < IMGW && xx >= 0 && xx < IMGW)
          v = mas[(size_t)bb_ * HW + yy * IMGW + xx];
        macc = fmaf(mw[ky * 3 + kx], v, macc);
      }
    const float msil = macc * sigm(macc);
    const float mm   = sigm(fmaf(msil, mw2, mb2));

    const float gate = om * s3a + mm * s3b + om * mm * s3c + s3d;

    // out[b,c,y,x] = cen[b,c,y,x] * gate — coalesced across lanes,
    // non-temporal stores (write-once stream; keep cen resident in L2 instead)
    const size_t base = (size_t)bb_ * 128 * HW + (size_t)(iy * IMGW + ix);
    #pragma unroll 8
    for (int c = 0; c < 128; ++c) {
      const size_t idx = base + (size_t)c * HW;
      __builtin_nontemporal_store(cen[idx] * gate, &out[idx]);
    }
  }
}

extern "C" void kernel_launch(void* const* d_in, const int* in_sizes, int n_in,
                              void* d_out, int out_size, void* d_ws, size_t ws_size,
                              hipStream_t stream) {
  (void)in_sizes; (void)n_in; (void)d_ws; (void)ws_size; (void)out_size;
  const float* cen    = (const float*)d_in[0];
  const float* mas    = (const float*)d_in[1];
  const float* w_in   = (const float*)d_in[2];
  const float* b_in   = (const float*)d_in[3];
  const float* scale1 = (const float*)d_in[4];
  const float* scale2 = (const float*)d_in[5];
  const float* scale3 = (const float*)d_in[6];
  const float* bc_w   = (const float*)d_in[7];
  const float* bn_g   = (const float*)d_in[8];
  const float* bn_b   = (const float*)d_in[9];
  const float* bn_m   = (const float*)d_in[10];
  const float* bn_v   = (const float*)d_in[11];
  const float* w_out  = (const float*)d_in[12];
  const float* b_out  = (const float*)d_in[13];
  const float* mas_w1 = (const float*)d_in[14];
  const float* mas_b1 = (const float*)d_in[15];
  const float* mas_w2 = (const float*)d_in[16];
  const float* mas_b2 = (const float*)d_in[17];
  float* out = (float*)d_out;

  dim3 grid(IMGW / TILE, IMGW / TILE, 4);
  dim3 block(256);
  ecm_fused<<<grid, block, 0, stream>>>(cen, mas, w_in, b_in, scale1, scale2, scale3,
                                        bc_w, bn_g, bn_b, bn_m, bn_v, w_out, b_out,
                                        mas_w1, mas_b1, mas_w2, mas_b2, out);
}